// EMD_distance_7438883356927
// MI455X (gfx1250) — compile-verified
//
#include <hip/hip_runtime.h>
#include <math.h>

typedef __attribute__((ext_vector_type(2))) float v2f;
typedef __attribute__((ext_vector_type(8))) float v8f;
typedef __attribute__((ext_vector_type(4))) int   v4i;

#define NB   8
#define NPTS 2048
#define TILE 16
#define WAVES_PER_BLOCK 8
#define BLOCK_THREADS   (WAVES_PER_BLOCK * 32)   // 256
#define ROWS_PER_BLOCK  (WAVES_PER_BLOCK * 16)   // 128
#define ROWBLOCKS       (NPTS / ROWS_PER_BLOCK)  // 16
#define PTS_PER_THREAD  (NPTS / BLOCK_THREADS)   // 8

#if __has_builtin(__builtin_amdgcn_global_load_async_to_lds_b128) && \
    __has_builtin(__builtin_amdgcn_global_load_async_to_lds_b32)  && \
    __has_builtin(__builtin_amdgcn_s_wait_asynccnt)
#define USE_ASYNC_LDS 1
#else
#define USE_ASYNC_LDS 0
#endif

#define AS_GLOBAL __attribute__((address_space(1)))
#define AS_LDS    __attribute__((address_space(3)))

static __device__ __forceinline__ float fast_exp2(float x) {
  return __builtin_amdgcn_exp2f(x);
}

#if USE_ASYNC_LDS
static __device__ __forceinline__ void async_cp16(void* lds, const void* g) {
  __builtin_amdgcn_global_load_async_to_lds_b128(
      (AS_GLOBAL v4i*)g, (AS_LDS v4i*)lds, 0, 0);
}
static __device__ __forceinline__ void async_cp4(void* lds, const void* g) {
  __builtin_amdgcn_global_load_async_to_lds_b32(
      (AS_GLOBAL int*)g, (AS_LDS int*)lds, 0, 0);
}
#endif

// ---------------------------------------------------------------------------
// init: pack (x,y,z,|p|^2) per point, init remain vectors / colsum / cost
// ---------------------------------------------------------------------------
__global__ void emd_init(const float* __restrict__ xyz1,
                         const float* __restrict__ xyz2,
                         float* __restrict__ pts4_1, float* __restrict__ pts4_2,
                         float* __restrict__ remainL, float* __restrict__ remainR,
                         float* __restrict__ colsumW, float* __restrict__ cost) {
  int idx = blockIdx.x * blockDim.x + threadIdx.x;
  if (idx < NB) cost[idx] = 0.f;
  if (idx >= NB * NPTS) return;
  {
    float x = xyz1[idx * 3 + 0], y = xyz1[idx * 3 + 1], z = xyz1[idx * 3 + 2];
    pts4_1[idx * 4 + 0] = x; pts4_1[idx * 4 + 1] = y;
    pts4_1[idx * 4 + 2] = z; pts4_1[idx * 4 + 3] = x * x + y * y + z * z;
  }
  {
    float x = xyz2[idx * 3 + 0], y = xyz2[idx * 3 + 1], z = xyz2[idx * 3 + 2];
    pts4_2[idx * 4 + 0] = x; pts4_2[idx * 4 + 1] = y;
    pts4_2[idx * 4 + 2] = z; pts4_2[idx * 4 + 3] = x * x + y * y + z * z;
  }
  remainL[idx] = 1.0f;   // n == m -> multiL = multiR = 1
  remainR[idx] = 1.0f;
  colsumW[idx] = 0.0f;
}

// ---------------------------------------------------------------------------
// ratio pass: ratioA[r] = remainA[r] / (sum_c exp(level*d2[r,c])*colW[c] + 1e-9)
// B operand is pre-scaled by k2 = level*log2(e) in LDS, so the WMMA directly
// yields k2*(-2 p.q + |q|^2); exp argument is a single v_add with k2*|p|^2.
// ---------------------------------------------------------------------------
__global__ __launch_bounds__(BLOCK_THREADS)
void emd_ratio_pass(const float* __restrict__ pts4A,
                    const float* __restrict__ pts4B,
                    const float* __restrict__ colW,
                    const float* __restrict__ remainA,
                    float* __restrict__ ratioA,
                    float level) {
  __shared__ float4 sB4[NPTS];        // raw stage, then WMMA-B operand pairs
  __shared__ float  sW[NPTS];

  const int b = blockIdx.y;
  const float k2  = level * 1.4426950408889634f;   // level * log2(e)
  const float km2 = -2.0f * k2;
  const float4* gB = (const float4*)(pts4B + (size_t)b * NPTS * 4);
  const float*  gW = colW + b * NPTS;
  const float4* gA = (const float4*)(pts4A + (size_t)b * NPTS * 4);
  const int tid = threadIdx.x;
  const int rowbase0 = blockIdx.x * ROWS_PER_BLOCK;
  __builtin_prefetch(gA + rowbase0 + (tid & 127), 0, 0);   // global_prefetch_b8

  float4 qreg[PTS_PER_THREAD];
#if USE_ASYNC_LDS
  // raw async fill (ASYNCcnt-tracked), then stage to registers
#pragma unroll
  for (int k = 0; k < PTS_PER_THREAD; ++k) {
    int i = tid + k * BLOCK_THREADS;
    async_cp16(&sB4[i], &gB[i]);
    async_cp4(&sW[i], &gW[i]);
  }
  __builtin_amdgcn_s_wait_asynccnt(0);
  __syncthreads();
#pragma unroll
  for (int k = 0; k < PTS_PER_THREAD; ++k) qreg[k] = sB4[tid + k * BLOCK_THREADS];
  __syncthreads();
#else
#pragma unroll
  for (int k = 0; k < PTS_PER_THREAD; ++k) {
    int i = tid + k * BLOCK_THREADS;
    qreg[k] = gB[i];
    sW[i] = gW[i];
  }
#endif
  // in-place transform: point i -> { km2*x, km2*y } { km2*z, k2*|q|^2 }
  float2* sB2 = (float2*)sB4;
#pragma unroll
  for (int k = 0; k < PTS_PER_THREAD; ++k) {
    int i = tid + k * BLOCK_THREADS;
    float4 q = qreg[k];
    sB2[2 * i + 0] = make_float2(km2 * q.x, km2 * q.y);
    sB2[2 * i + 1] = make_float2(km2 * q.z, k2 * q.w);
  }
  __syncthreads();

  const int wave = tid >> 5;
  const int lane = tid & 31;
  const int half = lane >> 4;        // K-pair select
  const int l16  = lane & 15;
  const int rowbase = rowbase0 + wave * 16;

  // A operand: lane l16 holds row (rowbase+l16); v0 = K0/K2, v1 = K1/K3
  float4 pA = gA[rowbase + l16];
  v2f a;
  a.x = half ? pA.z : pA.x;
  a.y = half ? 1.0f : pA.y;

  // k2*|p|^2 for the 8 rows this lane accumulates (VGPR v -> M = v + 8*half)
  float n1k[8];
#pragma unroll
  for (int v = 0; v < 8; ++v) n1k[v] = k2 * gA[rowbase + v + 8 * half].w;

  const float2* myB = sB2 + half;    // lane's K-pair
  float acc[8] = {0, 0, 0, 0, 0, 0, 0, 0};

  for (int t = 0; t < NPTS / TILE; ++t) {
    const int col = t * TILE + l16;
    float2 bq = myB[2 * col];        // single ds_load_b64, no VALU fixup
    v2f bv; bv.x = bq.x; bv.y = bq.y;
    v8f c = {};
    c = __builtin_amdgcn_wmma_f32_16x16x4_f32(
        false, a, false, bv, (short)0, c, false, false);
    const float wcol = sW[col];
#pragma unroll
    for (int v = 0; v < 8; ++v) {
      float e = fast_exp2(c[v] + n1k[v]);    // add + v_exp_f32
      acc[v]  = fmaf(e, wcol, acc[v]);
    }
  }

  // reduce each row's partial over the 16 column lanes of its half-wave
#pragma unroll
  for (int v = 0; v < 8; ++v) {
    float s = acc[v];
    for (int off = 8; off >= 1; off >>= 1) s += __shfl_xor(s, off, 32);
    if (l16 == 0) {
      int row = rowbase + v + 8 * half;
      ratioA[b * NPTS + row] = remainA[b * NPTS + row] / (s + 1e-9f);
    }
  }
}

// ---------------------------------------------------------------------------
// accumulate pass:  w = e * ratioL[r] * ratioR[c]
//   cost[b]    += sum w * sqrt(max(d2,1e-12))
//   remainL[r]  = max(0, remainL[r] - sum_c w)
//   colsumW[c] += sum_r w      (ds_add_f32 partials + one global atomic/col)
// B operand unscaled here because sqrt needs true d2.
// ---------------------------------------------------------------------------
__global__ __launch_bounds__(BLOCK_THREADS)
void emd_accum_pass(const float* __restrict__ pts4A,   // xyz1 side (rows)
                    const float* __restrict__ pts4B,   // xyz2 side (cols)
                    const float* __restrict__ ratioL,
                    const float* __restrict__ ratioR,
                    float* __restrict__ remainL,
                    float* __restrict__ colsumW,
                    float* __restrict__ cost,
                    float level) {
  __shared__ float4 sB4[NPTS];
  __shared__ float  sW[NPTS];    // ratioR
  __shared__ float  sCol[NPTS];  // per-column partial of sum_r w

  const int b = blockIdx.y;
  const float k2 = level * 1.4426950408889634f;
  const float4* gB = (const float4*)(pts4B + (size_t)b * NPTS * 4);
  const float*  gW = ratioR + b * NPTS;
  const float4* gA = (const float4*)(pts4A + (size_t)b * NPTS * 4);
  const int tid = threadIdx.x;
  const int rowbase0 = blockIdx.x * ROWS_PER_BLOCK;
  __builtin_prefetch(gA + rowbase0 + (tid & 127), 0, 0);

  float4 qreg[PTS_PER_THREAD];
#if USE_ASYNC_LDS
#pragma unroll
  for (int k = 0; k < PTS_PER_THREAD; ++k) {
    int i = tid + k * BLOCK_THREADS;
    async_cp16(&sB4[i], &gB[i]);
    async_cp4(&sW[i], &gW[i]);
  }
  __builtin_amdgcn_s_wait_asynccnt(0);
  __syncthreads();
#pragma unroll
  for (int k = 0; k < PTS_PER_THREAD; ++k) qreg[k] = sB4[tid + k * BLOCK_THREADS];
  __syncthreads();
#else
#pragma unroll
  for (int k = 0; k < PTS_PER_THREAD; ++k) {
    int i = tid + k * BLOCK_THREADS;
    qreg[k] = gB[i];
    sW[i] = gW[i];
  }
#endif
  float2* sB2 = (float2*)sB4;
#pragma unroll
  for (int k = 0; k < PTS_PER_THREAD; ++k) {
    int i = tid + k * BLOCK_THREADS;
    float4 q = qreg[k];
    sB2[2 * i + 0] = make_float2(-2.0f * q.x, -2.0f * q.y);
    sB2[2 * i + 1] = make_float2(-2.0f * q.z, q.w);
    sCol[i] = 0.0f;
  }
  __syncthreads();

  const int wave = tid >> 5;
  const int lane = tid & 31;
  const int half = lane >> 4;
  const int l16  = lane & 15;
  const int rowbase = rowbase0 + wave * 16;

  float4 pA = gA[rowbase + l16];
  v2f a;
  a.x = half ? pA.z : pA.x;
  a.y = half ? 1.0f : pA.y;

  float n1v[8], n1k[8], rlv[8];
#pragma unroll
  for (int v = 0; v < 8; ++v) {
    int row = rowbase + v + 8 * half;
    n1v[v] = gA[row].w;
    n1k[v] = k2 * n1v[v];
    rlv[v] = ratioL[b * NPTS + row];
  }

  const float2* myB = sB2 + half;
  float rowacc[8] = {0, 0, 0, 0, 0, 0, 0, 0};  // sum_c e * ratioR[c]
  float costacc = 0.0f;

  for (int t = 0; t < NPTS / TILE; ++t) {
    const int col = t * TILE + l16;
    float2 bq = myB[2 * col];
    v2f bv; bv.x = bq.x; bv.y = bq.y;
    v8f c = {};
    c = __builtin_amdgcn_wmma_f32_16x16x4_f32(
        false, a, false, bv, (short)0, c, false, false);
    const float rr = sW[col];
    float colacc = 0.0f;  // sum_v e * ratioL[row_v] for this lane's column
#pragma unroll
    for (int v = 0; v < 8; ++v) {
      float cross = c[v];                        // -2 p.q + |q|^2
      float d2 = cross + n1v[v];
      float e  = fast_exp2(fmaf(k2, cross, n1k[v]));   // exp2(k2*d2)
      float er = e * rlv[v];
      rowacc[v] = fmaf(e, rr, rowacc[v]);
      colacc   += er;
      costacc   = fmaf(er * rr, sqrtf(fmaxf(d2, 1e-12f)), costacc);
    }
    atomicAdd(&sCol[col], colacc * rr);   // ds_add_f32
  }

  // remainL update: one owner (lane 0 / lane 16) per row
#pragma unroll
  for (int v = 0; v < 8; ++v) {
    float s = rowacc[v];
    for (int off = 8; off >= 1; off >>= 1) s += __shfl_xor(s, off, 32);
    if (l16 == 0) {
      int row = rowbase + v + 8 * half;
      float rsum = rlv[v] * s;
      remainL[b * NPTS + row] = fmaxf(0.0f, remainL[b * NPTS + row] - rsum);
    }
  }

  // cost: reduce across full wave, one atomic per wave
  for (int off = 16; off >= 1; off >>= 1) costacc += __shfl_xor(costacc, off, 32);
  if (lane == 0) atomicAdd(&cost[b], costacc);

  __syncthreads();
  for (int i = tid; i < NPTS; i += BLOCK_THREADS)
    atomicAdd(&colsumW[b * NPTS + i], sCol[i]);
}

// ---------------------------------------------------------------------------
__global__ void emd_update(float* __restrict__ remainR, float* __restrict__ colsumW) {
  int idx = blockIdx.x * blockDim.x + threadIdx.x;
  if (idx >= NB * NPTS) return;
  remainR[idx] = fmaxf(0.0f, remainR[idx] - colsumW[idx]);
  colsumW[idx] = 0.0f;  // ready for next level
}

__global__ void emd_final(const float* __restrict__ cost, float* __restrict__ out) {
  int b = threadIdx.x;
  if (b < NB) out[b] = cost[b] * (1.0f / (float)NPTS);
}

// ---------------------------------------------------------------------------
extern "C" void kernel_launch(void* const* d_in, const int* in_sizes, int n_in,
                              void* d_out, int out_size, void* d_ws, size_t ws_size,
                              hipStream_t stream) {
  (void)in_sizes; (void)n_in; (void)out_size; (void)ws_size;
  const float* xyz1 = (const float*)d_in[0];
  const float* xyz2 = (const float*)d_in[1];
  float* out = (float*)d_out;

  float* ws      = (float*)d_ws;
  float* pts4_1  = ws;                         // NB*NPTS*4
  float* pts4_2  = pts4_1 + NB * NPTS * 4;     // NB*NPTS*4
  float* remainL = pts4_2 + NB * NPTS * 4;     // NB*NPTS
  float* remainR = remainL + NB * NPTS;
  float* ratioL  = remainR + NB * NPTS;
  float* ratioR  = ratioL + NB * NPTS;
  float* colsumW = ratioR + NB * NPTS;
  float* cost    = colsumW + NB * NPTS;        // NB

  const int elems = NB * NPTS;
  const int tblocks = (elems + 255) / 256;

  emd_init<<<tblocks, 256, 0, stream>>>(xyz1, xyz2, pts4_1, pts4_2,
                                        remainL, remainR, colsumW, cost);

  // levels: -(4^j) for j = 7..-1, then 0
  const float levels[10] = {-16384.f, -4096.f, -1024.f, -256.f, -64.f,
                            -16.f,    -4.f,    -1.f,    -0.25f, 0.f};

  dim3 grid(ROWBLOCKS, NB);
  for (int l = 0; l < 10; ++l) {
    const float lv = levels[l];
    // pass 1: rows = xyz1, col weights = remainR  ->  ratioL
    emd_ratio_pass<<<grid, BLOCK_THREADS, 0, stream>>>(
        pts4_1, pts4_2, remainR, remainL, ratioL, lv);
    // pass 2: rows = xyz2, col weights = ratioL   ->  ratioR
    emd_ratio_pass<<<grid, BLOCK_THREADS, 0, stream>>>(
        pts4_2, pts4_1, ratioL, remainR, ratioR, lv);
    // pass 3: accumulate w into cost / remainL / colsumW
    emd_accum_pass<<<grid, BLOCK_THREADS, 0, stream>>>(
        pts4_1, pts4_2, ratioL, ratioR, remainL, colsumW, cost, lv);
    // remainR update + colsum reset
    emd_update<<<tblocks, 256, 0, stream>>>(remainR, colsumW);
  }

  emd_final<<<1, NB, 0, stream>>>(cost, out);
}